// StabilityLossCoP_63634235457554
// MI455X (gfx1250) — compile-verified
//
#include <hip/hip_runtime.h>
#include <math.h>

// Problem constants (match reference)
#define NUM_PARTS 10
#define NB   512
#define NV   6890
#define NF   13776
#define NVHD 20000
#define COS_W 10.0f
#define COS_K 100.0f
#define EPSV  1e-6f

#define BLOCK_T 512
#define NWAVES  (BLOCK_T / 32)

// ---- CDNA5 async global->LDS path (guarded; fallback is synchronous) ----
#if defined(__HIP_DEVICE_COMPILE__) && defined(__gfx1250__)
#if __has_builtin(__builtin_amdgcn_global_load_async_to_lds_b32) && \
    __has_builtin(__builtin_amdgcn_s_wait_asynccnt)
#define USE_ASYNC_LDS 1
#endif
#endif
#ifndef USE_ASYNC_LDS
#define USE_ASYNC_LDS 0
#endif

__device__ __forceinline__ float wave_reduce_sum(float v) {
#pragma unroll
    for (int off = 16; off > 0; off >>= 1)
        v += __shfl_xor(v, off, 32);
    return v;
}

__launch_bounds__(BLOCK_T, 1)
__global__ void StabilityLossCoP_kernel(const float* __restrict__ vertices,  // (B, V, 3)
                                        const float* __restrict__ bary,      // (VHD, 3)
                                        const int*   __restrict__ faces,     // (F, 3)
                                        const int*   __restrict__ hd_fid,    // (VHD,)
                                        const int*   __restrict__ part_fid,  // (F,)
                                        float*       __restrict__ out)       // (B,)
{
    // Padded float4 layout: every vertex gather is one aligned ds_load_b128.
    __shared__ float4 verts4[NV];                  // 110240 B
    __shared__ float partVol[NUM_PARTS];
    __shared__ float redPart[NWAVES][NUM_PARTS];
    __shared__ float red6[NWAVES][6];

    const int b    = blockIdx.x;
    const int tid  = threadIdx.x;
    const int wave = tid >> 5;
    const int lane = tid & 31;
    const float* __restrict__ vbase = vertices + (size_t)b * (NV * 3);

    // ---- Stage vertices[b] into padded LDS slots (async global->LDS, ASYNCcnt) ----
#if USE_ASYNC_LDS
    for (int v = tid; v < NV; v += BLOCK_T) {
        float* dst = (float*)&verts4[v];
#pragma unroll
        for (int k = 0; k < 3; ++k) {
            __builtin_amdgcn_global_load_async_to_lds_b32(
                (__attribute__((address_space(1))) int*)(vbase + 3 * v + k),
                (__attribute__((address_space(3))) int*)(dst + k),
                /*offset=*/0, /*cpol=*/0);
        }
    }
    __builtin_amdgcn_s_wait_asynccnt(0);
#else
    for (int v = tid; v < NV; v += BLOCK_T) {
        float4 t;
        t.x = vbase[3 * v + 0];
        t.y = vbase[3 * v + 1];
        t.z = vbase[3 * v + 2];
        t.w = 0.0f;
        verts4[v] = t;
    }
#endif
    __syncthreads();

    // ---- Phase A: per-part signed volumes (deterministic reduction) ----
    float pv[NUM_PARTS];
#pragma unroll
    for (int p = 0; p < NUM_PARTS; ++p) pv[p] = 0.0f;

    for (int f = tid; f < NF; f += BLOCK_T) {
        const int i0 = faces[3 * f + 0];
        const int i1 = faces[3 * f + 1];
        const int i2 = faces[3 * f + 2];
        const float4 A = verts4[i0];   // ds_load_b128
        const float4 Bv = verts4[i1];
        const float4 C = verts4[i2];
        // dot(cross(t0, t1), t2) / 6
        const float crx = A.y * Bv.z - A.z * Bv.y;
        const float cry = A.z * Bv.x - A.x * Bv.z;
        const float crz = A.x * Bv.y - A.y * Bv.x;
        const float vol = (crx * C.x + cry * C.y + crz * C.z) * (1.0f / 6.0f);
        const int part = part_fid[f];
#pragma unroll
        for (int p = 0; p < NUM_PARTS; ++p)
            pv[p] += (p == part) ? vol : 0.0f;
    }
#pragma unroll
    for (int p = 0; p < NUM_PARTS; ++p) {
        const float s = wave_reduce_sum(pv[p]);
        if (lane == 0) redPart[wave][p] = s;
    }
    __syncthreads();
    if (tid < NUM_PARTS) {
        float s = 0.0f;
        for (int w = 0; w < NWAVES; ++w) s += redPart[w][tid];
        partVol[tid] = s;
    }
    __syncthreads();

    // ---- Phase B: HD vertices, weighted sums (y-sums dead: out uses x,z only) ----
    float sxv = 0.0f, szv = 0.0f, sv = 0.0f;   // com numerators / denom
    float sxp = 0.0f, szp = 0.0f, sp = 0.0f;   // cop numerators / denom

    for (int v = tid; v < NVHD; v += BLOCK_T) {
        const int fid = hd_fid[v];
        const int i0 = faces[3 * fid + 0];
        const int i1 = faces[3 * fid + 1];
        const int i2 = faces[3 * fid + 2];
        const float b0 = bary[3 * v + 0], b1 = bary[3 * v + 1], b2 = bary[3 * v + 2];
        const float inv = 1.0f / (b0 + b1 + b2);
        const float w0 = b0 * inv, w1 = b1 * inv, w2 = b2 * inv;

        const float4 A = verts4[i0];   // ds_load_b128
        const float4 Bv = verts4[i1];
        const float4 C = verts4[i2];

        const float x = w0 * A.x + w1 * Bv.x + w2 * C.x;
        const float y = w0 * A.y + w1 * Bv.y + w2 * C.y;
        const float z = w0 * A.z + w1 * Bv.z + w2 * C.z;

        const float vol = partVol[part_fid[fid]];
        sxv += x * vol;  szv += z * vol;  sv += vol;

        const float pw = (y < 0.0f) ? fmaf(-COS_K, y, 1.0f) : expf(-COS_W * y);
        sxp += x * pw;   szp += z * pw;   sp += pw;
    }

    float acc[6] = { sxv, szv, sv, sxp, szp, sp };
#pragma unroll
    for (int k = 0; k < 6; ++k) {
        const float s = wave_reduce_sum(acc[k]);
        if (lane == 0) red6[wave][k] = s;
    }
    __syncthreads();
    if (tid == 0) {
        float t[6] = { 0, 0, 0, 0, 0, 0 };
        for (int w = 0; w < NWAVES; ++w)
#pragma unroll
            for (int k = 0; k < 6; ++k) t[k] += red6[w][k];
        const float comx = t[0] / t[2];
        const float comz = t[1] / t[2];
        const float copx = t[3] / (t[5] + EPSV);
        const float copz = t[4] / (t[5] + EPSV);
        const float dx = comx - copx;
        const float dz = comz - copz;
        out[b] = sqrtf(dx * dx + dz * dz);
    }
}

extern "C" void kernel_launch(void* const* d_in, const int* in_sizes, int n_in,
                              void* d_out, int out_size, void* d_ws, size_t ws_size,
                              hipStream_t stream) {
    (void)in_sizes; (void)n_in; (void)d_ws; (void)ws_size; (void)out_size;
    const float* vertices = (const float*)d_in[0];
    const float* bary     = (const float*)d_in[1];
    const int*   faces    = (const int*)d_in[2];
    const int*   hd_fid   = (const int*)d_in[3];
    const int*   part_fid = (const int*)d_in[4];
    float* out = (float*)d_out;

    StabilityLossCoP_kernel<<<dim3(NB), dim3(BLOCK_T), 0, stream>>>(
        vertices, bary, faces, hd_fid, part_fid, out);
}